// QuantumLayer_44968307589412
// MI455X (gfx1250) — compile-verified
//
#include <hip/hip_runtime.h>
#include <math.h>

typedef __attribute__((ext_vector_type(2))) float v2f;
typedef __attribute__((ext_vector_type(8))) float v8f;

#define DEVINL __device__ __forceinline__

// Natural-row -> A-slot permutation so that WMMA D output lands directly in
// B-operand layout: slot m holds natural row perm(m).
DEVINL int permrow(int m) { return 4 * ((m & 7) >> 1) + 2 * (m >> 3) + (m & 1); }

// Gate ids:
// 0 = H (x) H (x) H (x) H   (folded initial layer, entries +-0.25)
// 1 = H(q0/bit8)  2 = X(q1/bit4)  3 = Y(q2/bit2) [pure imag, stores Im(U)]
// 4 = Z(q3/bit1)  5 = CNOT(q0->q1) 6 = CZ(q1,q2) 7 = TOF(q2,q3->q0)
// 8 = CSWAP(q3 ctrl; swap q0,q1)
DEVINL float gate_entry(int g, int o, int i) {
    switch (g) {
    case 0: return 0.25f * ((__popc(o & i) & 1) ? -1.0f : 1.0f);
    case 1: return ((o & 7) == (i & 7))
                 ? (0.70710678118654752f * (((o & 8) && (i & 8)) ? -1.0f : 1.0f))
                 : 0.0f;
    case 2: return (o == (i ^ 4)) ? 1.0f : 0.0f;
    case 3: return (o == (i ^ 2)) ? ((i & 2) ? -1.0f : 1.0f) : 0.0f; // Im(Y2)
    case 4: return (o == i) ? ((i & 1) ? -1.0f : 1.0f) : 0.0f;
    case 5: { int t = (i & 8) ? (i ^ 4) : i; return (o == t) ? 1.0f : 0.0f; }
    case 6: return (o == i) ? (((i & 4) && (i & 2)) ? -1.0f : 1.0f) : 0.0f;
    case 7: { int t = ((i & 2) && (i & 1)) ? (i ^ 8) : i; return (o == t) ? 1.0f : 0.0f; }
    default: { int t = (i & 1) ? ((i & ~12) | ((i >> 1) & 4) | ((i << 1) & 8)) : i;
               return (o == t) ? 1.0f : 0.0f; }
    }
}

// Pure-VALU swap of the two 16-lane halves of the wave32
// (v_permlanex16_b32 with identity lane selects).
DEVINL float swap_half(float v) {
    int iv = __float_as_int(v);
    int r = __builtin_amdgcn_permlanex16(iv, iv, 0x76543210, 0xfedcba98, true, false);
    return __int_as_float(r);
}

// D = Mg_perm * S : state S is already in B-operand layout, so chunk k's
// B operand is simply {S[2k], S[2k+1]} -- no shuffles, no selects.
DEVINL v8f matmul16(const float* Mg, const v8f& S, int row, int hi) {
    v8f acc = {};
#pragma unroll
    for (int k = 0; k < 4; ++k) {
        const v2f a = *reinterpret_cast<const v2f*>(Mg + row * 16 + 4 * k + 2 * hi);
        v2f b; b[0] = S[2 * k]; b[1] = S[2 * k + 1];
        acc = __builtin_amdgcn_wmma_f32_16x16x4_f32(
            false, a, false, b, (short)0, acc, false, false);
    }
    return acc;
}

DEVINL void apply_fixed(const float* sM, int g, bool isImag,
                        v8f& Sr, v8f& Si, int row, int hi) {
    const float* Mg = sM + g * 256;
    if (!isImag) {
        v8f nr = matmul16(Mg, Sr, row, hi);
        v8f ni = matmul16(Mg, Si, row, hi);
        Sr = nr; Si = ni;
    } else {
        // U = i*M (M real):  Sr' = -M*Si ; Si' = M*Sr
        v8f t  = matmul16(Mg, Si, row, hi);
        v8f ni = matmul16(Mg, Sr, row, hi);
#pragma unroll
        for (int v = 0; v < 8; ++v) Sr[v] = -t[v];
        Si = ni;
    }
}

// Rotations on the B-layout state.
// Natural row r sits at (VGPR p = 2*(r>>2) + (r&1), half h = (r>>1)&1), so:
//   qubit0 (bit8): in-lane pairs (p, p+4)
//   qubit1 (bit4): in-lane pairs (p, p+2)
//   qubit3 (bit1): in-lane pairs (p, p+1)
//   qubit2 (bit2): cross-half (lane ^ 16, same VGPR)
// kind: 0=Rx, 1=Ry, 2=Rz ; c=cos(th/2), s=sin(th/2).
DEVINL void apply_rot(int kind, int q, float c, float s,
                      v8f& Sr, v8f& Si, int hi) {
    if (q == 2) {
        if (kind == 0) {            // Rx: nr = c*r + s*partner_i ; ni = c*i - s*partner_r
#pragma unroll
            for (int v = 0; v < 8; ++v) {
                float pr = swap_half(Sr[v]), pi = swap_half(Si[v]);
                float nr = c * Sr[v] + s * pi;
                float ni = c * Si[v] - s * pr;
                Sr[v] = nr; Si[v] = ni;
            }
        } else if (kind == 1) {     // Ry
            float sp = hi ? s : -s;
#pragma unroll
            for (int v = 0; v < 8; ++v) {
                float pr = swap_half(Sr[v]), pi = swap_half(Si[v]);
                Sr[v] = c * Sr[v] + sp * pr;
                Si[v] = c * Si[v] + sp * pi;
            }
        } else {                    // Rz (diagonal)
            float sp = hi ? -s : s;
#pragma unroll
            for (int v = 0; v < 8; ++v) {
                float nr = c * Sr[v] + sp * Si[v];
                float ni = c * Si[v] - sp * Sr[v];
                Sr[v] = nr; Si[v] = ni;
            }
        }
    } else {
        const int d = (q == 0) ? 4 : (q == 1) ? 2 : 1;
#pragma unroll
        for (int v0 = 0; v0 < 8; ++v0) {
            if (v0 & d) continue;
            const int v1 = v0 + d;
            float r0 = Sr[v0], i0 = Si[v0], r1 = Sr[v1], i1 = Si[v1];
            if (kind == 0) {        // Rx
                Sr[v0] = c * r0 + s * i1;
                Si[v0] = c * i0 - s * r1;
                Sr[v1] = c * r1 + s * i0;
                Si[v1] = c * i1 - s * r0;
            } else if (kind == 1) { // Ry
                Sr[v0] = c * r0 - s * r1;
                Si[v0] = c * i0 - s * i1;
                Sr[v1] = c * r1 + s * r0;
                Si[v1] = c * i1 + s * i0;
            } else {                // Rz
                Sr[v0] = c * r0 + s * i0;
                Si[v0] = c * i0 - s * r0;
                Sr[v1] = c * r1 - s * i1;
                Si[v1] = c * i1 + s * r1;
            }
        }
    }
}

__launch_bounds__(256, 2)
__global__ void qlayer_kernel(const float* __restrict__ x1,
                              const float* __restrict__ x2,
                              const float* __restrict__ qp,
                              float* __restrict__ out, int B) {
    __shared__ float sM[9 * 256];
    {
        const int t = threadIdx.x;      // 256 threads: one (slot-row, col) pair each
        const int o = t >> 4, i = t & 15;
        const int po = permrow(o);      // A slot o holds natural row perm(o)
#pragma unroll
        for (int g = 0; g < 9; ++g) sM[g * 256 + o * 16 + i] = gate_entry(g, po, i);
    }
    __syncthreads();

    const int lane = threadIdx.x & 31;
    const int wv   = threadIdx.x >> 5;
    const int hi   = lane >> 4;
    const int row  = lane & 15;         // A-matrix slot row AND batch column
    const int b    = (blockIdx.x * 8 + wv) * 16 + row;
    const int bc   = b < B ? b : (B - 1);

    float xa[4], xb[4];
#pragma unroll
    for (int n = 0; n < 4; ++n) { xa[n] = x1[bc * 4 + n]; xb[n] = x2[bc * 4 + n]; }
    float w[24];
#pragma unroll
    for (int j = 0; j < 24; ++j) w[j] = qp[j];

    // state = Hall * |0> : every natural-row amplitude = 0.25 (layout-invariant)
    v8f Sr, Si;
#pragma unroll
    for (int v = 0; v < 8; ++v) { Sr[v] = 0.25f; Si[v] = 0.0f; }

    // ---------------- forward pass ----------------
#pragma unroll
    for (int l = 0; l < 6; ++l) {
#pragma unroll
        for (int n = 0; n < 4; ++n) {
            if (((l + n) & 1) == 0) {                  // Ry(x1[:,n]) on qubit n
                float s, c; __sincosf(0.5f * xa[n], &s, &c);
                apply_rot(1, n, c, s, Sr, Si, hi);
            }
            {                                          // R_kind(w[l,n]) on qubit n
                float s, c; __sincosf(0.5f * w[l * 4 + n], &s, &c);
                apply_rot((l + n) % 3, n, c, s, Sr, Si, hi);
            }
            const int t = (l * 4 + n) & 7;             // fixed gate
            apply_fixed(sM, t + 1, t == 2, Sr, Si, row, hi);
        }
    }

    // ---------------- adjoint pass ----------------
#pragma unroll
    for (int l = 5; l >= 0; --l) {
#pragma unroll
        for (int n = 3; n >= 0; --n) {
            const int t = (l * 4 + n) & 7;             // fixed gates are self-adjoint
            apply_fixed(sM, t + 1, t == 2, Sr, Si, row, hi);
            {
                float s, c; __sincosf(-0.5f * w[l * 4 + n], &s, &c);
                apply_rot((l + n) % 3, n, c, s, Sr, Si, hi);
            }
            if (((l + n) & 1) == 0) {
                float s, c; __sincosf(-0.5f * xb[n], &s, &c);
                apply_rot(1, n, c, s, Sr, Si, hi);
            }
        }
    }
    // trailing (H (x) H (x) H (x) H) of the adjoint pass
    apply_fixed(sM, 0, false, Sr, Si, row, hi);

    // amplitude of natural row 0: slot p=0, half h=0 -> lanes 0..15, VGPR 0
    if (hi == 0 && b < B) {
        float ar = Sr[0], ai = Si[0];
        out[b] = ar * ar + ai * ai;
    }
}

extern "C" void kernel_launch(void* const* d_in, const int* in_sizes, int n_in,
                              void* d_out, int out_size, void* d_ws, size_t ws_size,
                              hipStream_t stream) {
    const float* x1 = (const float*)d_in[0];
    const float* x2 = (const float*)d_in[1];
    const float* qp = (const float*)d_in[2];
    float* out = (float*)d_out;
    const int B = out_size;                 // 262144
    const int elemsPerBlock = 128;          // 8 waves * 16 columns
    const int grid = (B + elemsPerBlock - 1) / elemsPerBlock;
    qlayer_kernel<<<grid, 256, 0, stream>>>(x1, x2, qp, out, B);
}